// RankingLossWithTies_59648505806944
// MI455X (gfx1250) — compile-verified
//
#include <hip/hip_runtime.h>

typedef float v2f __attribute__((ext_vector_type(2)));
typedef float v8f __attribute__((ext_vector_type(8)));

#ifndef __has_builtin
#define __has_builtin(x) 0
#endif
#if __has_builtin(__builtin_amdgcn_wmma_f32_16x16x4_f32)
#define HAVE_WMMA_F32X4 1
#else
#define HAVE_WMMA_F32X4 0
#endif

namespace {
constexpr int   kB           = 256;
constexpr int   kN           = 512;
constexpr int   kTile        = 16;
constexpr int   kTilesPerRow = kN / kTile;             // 32
constexpr int   kWaves       = 8;                      // 256 threads / wave32
constexpr int   kJIter       = kTilesPerRow / kWaves;  // 4 j-tiles per wave
constexpr float kMargin      = 1.0f;
// Diagonal pairs are NOT masked in the hot loop: each contributes exactly
// 0.0f to the loss (WMMA pd is exactly 0 on the diagonal) and exactly +1 to
// the count (ri==rj). We subtract them analytically in the finalize kernel.
constexpr unsigned int kDiagPairs = (unsigned int)kB * (unsigned int)kN;  // 131072
}

// d_ws layout: ws[0] = float loss accumulator, ((u32*)ws)[1] = exact pair count.
__global__ void rlt_zero_ws(float* ws) {
    ws[0] = 0.0f;
    reinterpret_cast<unsigned int*>(ws)[1] = 0u;
}

__global__ __launch_bounds__(256) void rlt_pair_kernel(
    const float* __restrict__ pred, const int* __restrict__ rank, float* __restrict__ ws)
{
    const int blk  = blockIdx.x;        // b*32 + i-tile
    const int b    = blk >> 5;
    const int it   = blk & 31;
    const int i0   = it * kTile;

    const int tid  = threadIdx.x;
    const int wave = tid >> 5;          // 0..7
    const int lane = tid & 31;
    const int half = lane >> 4;         // 0: K=0,1 / D rows M=0..7 ; 1: K=2,3 / D rows M=8..15
    const int nIdx = lane & 15;

    const float* prow = pred + b * kN;
    const int*   rrow = rank + b * kN;

    // A fragment (16x4 f32, ISA layout: lanes 0-15 -> M=lane, VGPR0=K0, VGPR1=K1;
    // lanes 16-31 -> M=lane-16, VGPR0=K2, VGPR1=K3). A columns = [p_i, 1, 0, 0].
    v2f afrag;
    afrag.x = (half == 0) ? prow[i0 + nIdx] : 0.0f;
    afrag.y = (half == 0) ? 1.0f            : 0.0f;

    // Ranks (and, for the fallback, preds) of this lane's 8 D rows: M = g + half*8.
    int ri[8];
#if !HAVE_WMMA_F32X4
    float pi[8];
#endif
#pragma unroll
    for (int g = 0; g < 8; ++g) {
        ri[g] = rrow[i0 + g + half * 8];
#if !HAVE_WMMA_F32X4
        pi[g] = prow[i0 + g + half * 8];
#endif
    }

    float        lossAcc = 0.0f;
    unsigned int cntAcc  = 0u;

#pragma unroll
    for (int u = 0; u < kJIter; ++u) {
        const int   jt = wave + u * kWaves;   // wave-uniform
        const int   j0 = jt * kTile;
        const int   rj = rrow[j0 + nIdx];
        const float pj = prow[j0 + nIdx];

        v8f d;
#if HAVE_WMMA_F32X4
        // B fragment (4x16 f32): rows = [1; -p_j; 0; 0]; N = lane%16, lane-half picks K pair.
        v2f bfrag;
        bfrag.x = (half == 0) ? 1.0f : 0.0f;
        bfrag.y = (half == 0) ? -pj  : 0.0f;
        v8f c = {};
        // D[m][n] = p_{i0+m}*1 + 1*(-p_{j0+n}) = exact f32 pairwise-difference tile
        // (products are x1 so no rounding; diagonal is exactly 0).
        d = __builtin_amdgcn_wmma_f32_16x16x4_f32(/*neg_a=*/false, afrag,
                                                  /*neg_b=*/false, bfrag,
                                                  /*c_mod=*/(short)0, c,
                                                  /*reuse_a=*/false, /*reuse_b=*/false);
#else
#pragma unroll
        for (int g = 0; g < 8; ++g) d[g] = pi[g] - pj;
#endif

        // Branch-free: 2 v_cmp + 2 v_cndmask + 3 f32 ops + 2 accumulates per element.
        // gt|eq == (ri >= rj); diagonal handled analytically (see kDiagPairs).
#pragma unroll
        for (int g = 0; g < 8; ++g) {
            const float pd    = d[g];
            const int   rg    = ri[g];
            const bool  ge    = (rg >= rj);
            const bool  gt    = (rg >  rj);
            const float hinge = fmaxf(kMargin + pd, 0.0f);
            const float sq    = pd * pd;
            lossAcc += ge ? (gt ? hinge : sq) : 0.0f;
            cntAcc  += ge ? 1u : 0u;
        }
    }

    // In-wave tree reduction (wave32).
#pragma unroll
    for (int off = 16; off > 0; off >>= 1) {
        lossAcc += __shfl_down(lossAcc, off, 32);
        cntAcc  += __shfl_down(cntAcc,  off, 32);
    }

    __shared__ float        sLoss[kWaves];
    __shared__ unsigned int sCnt[kWaves];
    if (lane == 0) { sLoss[wave] = lossAcc; sCnt[wave] = cntAcc; }
    __syncthreads();

    if (tid == 0) {
        float        L = 0.0f;
        unsigned int C = 0u;
#pragma unroll
        for (int w = 0; w < kWaves; ++w) { L += sLoss[w]; C += sCnt[w]; }
        atomicAdd(&ws[0], L);
        atomicAdd(reinterpret_cast<unsigned int*>(ws) + 1, C);
    }
}

__global__ void rlt_finalize(const float* __restrict__ ws, float* __restrict__ out) {
    const float loss = ws[0];
    const unsigned int rawCnt = reinterpret_cast<const unsigned int*>(ws)[1];
    const float cnt = (float)(rawCnt - kDiagPairs);   // remove the B*N diagonal pairs
    out[0] = loss / (cnt + 1e-8f);
}

extern "C" void kernel_launch(void* const* d_in, const int* in_sizes, int n_in,
                              void* d_out, int out_size, void* d_ws, size_t ws_size,
                              hipStream_t stream) {
    const float* pred = (const float*)d_in[0];
    const int*   rank = (const int*)d_in[1];  // integer input -> const int* per harness contract
    float* out = (float*)d_out;
    float* ws  = (float*)d_ws;

    rlt_zero_ws<<<1, 1, 0, stream>>>(ws);
    rlt_pair_kernel<<<kB * kTilesPerRow, 256, 0, stream>>>(pred, rank, ws);
    rlt_finalize<<<1, 1, 0, stream>>>(ws, out);
}